// TorchIEKF_83897891160758
// MI455X (gfx1250) — compile-verified
//
#include <hip/hip_runtime.h>

// ---------------------------------------------------------------------------
// IEKF sequential scan on one wave32, 21x21 covariance algebra padded to
// 32x32 and executed on the CDNA5 matrix pipe via V_WMMA_F32_16X16X4_F32.
// ---------------------------------------------------------------------------

typedef float v2f __attribute__((ext_vector_type(2)));
typedef float v8f __attribute__((ext_vector_type(8)));

#define MM(i, j) ((i) * 32 + (j))

// ---------------- small 3x3 helpers (per-lane registers, uniform) -----------
__device__ __forceinline__ void skew3(const float* x, float* S) {
  S[0] = 0.f;   S[1] = -x[2]; S[2] = x[1];
  S[3] = x[2];  S[4] = 0.f;   S[5] = -x[0];
  S[6] = -x[1]; S[7] = x[0];  S[8] = 0.f;
}

__device__ __forceinline__ void m3mul(const float* A, const float* B, float* C) {
#pragma unroll
  for (int i = 0; i < 3; ++i)
#pragma unroll
    for (int j = 0; j < 3; ++j)
      C[i * 3 + j] = A[i * 3 + 0] * B[0 * 3 + j] + A[i * 3 + 1] * B[1 * 3 + j] +
                     A[i * 3 + 2] * B[2 * 3 + j];
}

__device__ __forceinline__ void m3Tmul(const float* A, const float* B, float* C) {
  // C = A^T * B
#pragma unroll
  for (int i = 0; i < 3; ++i)
#pragma unroll
    for (int j = 0; j < 3; ++j)
      C[i * 3 + j] = A[0 * 3 + i] * B[0 * 3 + j] + A[1 * 3 + i] * B[1 * 3 + j] +
                     A[2 * 3 + i] * B[2 * 3 + j];
}

__device__ __forceinline__ void m3v(const float* A, const float* x, float* y) {
#pragma unroll
  for (int i = 0; i < 3; ++i)
    y[i] = A[i * 3 + 0] * x[0] + A[i * 3 + 1] * x[1] + A[i * 3 + 2] * x[2];
}

__device__ __forceinline__ void m3Tv(const float* A, const float* x, float* y) {
#pragma unroll
  for (int i = 0; i < 3; ++i)
    y[i] = A[0 * 3 + i] * x[0] + A[1 * 3 + i] * x[1] + A[2 * 3 + i] * x[2];
}

__device__ __forceinline__ void so3exp(const float* phi, float* R) {
  float sq = phi[0] * phi[0] + phi[1] * phi[1] + phi[2] * phi[2];
  if (sq < 1e-16f) {
    float S[9];
    skew3(phi, S);
#pragma unroll
    for (int k = 0; k < 9; ++k) R[k] = S[k];
    R[0] += 1.f; R[4] += 1.f; R[8] += 1.f;
  } else {
    float ang = sqrtf(sq);
    float ax[3] = {phi[0] / ang, phi[1] / ang, phi[2] / ang};
    float s = sinf(ang), c = cosf(ang);
    float S[9];
    skew3(ax, S);
#pragma unroll
    for (int i = 0; i < 3; ++i)
#pragma unroll
      for (int j = 0; j < 3; ++j) {
        float d = (i == j) ? 1.f : 0.f;
        R[i * 3 + j] = c * d + (1.f - c) * ax[i] * ax[j] + s * S[i * 3 + j];
      }
  }
}

__device__ __forceinline__ void sen3exp(const float* xi, float* R, float* x0, float* x1) {
  float sq = xi[0] * xi[0] + xi[1] * xi[1] + xi[2] * xi[2];
  float J[9];
  if (sq < 1e-16f) {
    float S[9];
    skew3(xi, S);
#pragma unroll
    for (int k = 0; k < 9; ++k) {
      R[k] = S[k];
      J[k] = 0.5f * S[k];
    }
    R[0] += 1.f; R[4] += 1.f; R[8] += 1.f;
    J[0] += 1.f; J[4] += 1.f; J[8] += 1.f;
  } else {
    float ang = sqrtf(sq);
    float ax[3] = {xi[0] / ang, xi[1] / ang, xi[2] / ang};
    float s = sinf(ang), c = cosf(ang);
    float sa = s / ang;
    float S[9];
    skew3(ax, S);
#pragma unroll
    for (int i = 0; i < 3; ++i)
#pragma unroll
      for (int j = 0; j < 3; ++j) {
        float o = ax[i] * ax[j];
        float d = (i == j) ? 1.f : 0.f;
        R[i * 3 + j] = c * d + (1.f - c) * o + s * S[i * 3 + j];
        J[i * 3 + j] = sa * d + (1.f - sa) * o + ((1.f - c) / ang) * S[i * 3 + j];
      }
  }
  m3v(J, xi + 3, x0);
  m3v(J, xi + 6, x1);
}

// ---------------- 32x32 f32 matmul on the WMMA pipe -------------------------
// D = A @ B, all row-major 32x32 in LDS. One wave32, EXEC all ones.
// Uses V_WMMA_F32_16X16X4_F32: A frag 16x4 (2 VGPR/lane), B frag 4x16,
// C/D 16x16 (8 VGPR/lane), per the CDNA5 ISA layouts.
__device__ __forceinline__ void mm32(const float* __restrict__ A,
                                     const float* __restrict__ B,
                                     float* __restrict__ D, int lane) {
  const int half = lane >> 4;  // 0: lanes 0-15, 1: lanes 16-31
  const int l16 = lane & 15;
#pragma unroll
  for (int ti = 0; ti < 2; ++ti) {
#pragma unroll
    for (int tj = 0; tj < 2; ++tj) {
      v8f acc = {};
#pragma unroll
      for (int k0 = 0; k0 < 8; ++k0) {
        const int kk = k0 * 4 + half * 2;  // K base for this lane-half
        v2f a, b;
        a.x = A[(ti * 16 + l16) * 32 + kk];      // A[M][kk]
        a.y = A[(ti * 16 + l16) * 32 + kk + 1];  // A[M][kk+1]
        b.x = B[kk * 32 + tj * 16 + l16];        // B[kk][N]
        b.y = B[(kk + 1) * 32 + tj * 16 + l16];  // B[kk+1][N]
        acc = __builtin_amdgcn_wmma_f32_16x16x4_f32(
            false, a, false, b, (short)0, acc, false, false);
      }
#pragma unroll
      for (int r = 0; r < 8; ++r) {
        // C/D layout: VGPR r -> M = r (lanes 0-15) or r+8 (lanes 16-31)
        D[(ti * 16 + r + half * 8) * 32 + tj * 16 + l16] = acc[r];
      }
    }
  }
}

// ---------------------------------------------------------------------------
__global__ __launch_bounds__(32) void iekf_scan_kernel(
    const float* __restrict__ t, const float* __restrict__ u,
    const float* __restrict__ mc, const float* __restrict__ vmes,
    const float* __restrict__ ang0, float* __restrict__ out, int N) {
  __shared__ float Pl[1024];
  __shared__ float bufA[1024];
  __shared__ float bufB[1024];
  __shared__ float bufC[1024];
  __shared__ float bufD[1024];
  __shared__ float Hld[2][21];
  __shared__ float PHtld[21][2];
  __shared__ float Kld[21][2];
  __shared__ float dxld[21];

  const int lane = threadIdx.x;

  // output layout: Rot | v | p | b_omega | b_acc | Rot_c_i | t_c_i
  float* oRot = out;
  float* oV   = out + (size_t)N * 9;
  float* oP   = out + (size_t)N * 12;
  float* oBo  = out + (size_t)N * 15;
  float* oBa  = out + (size_t)N * 18;
  float* oRci = out + (size_t)N * 21;
  float* oTci = out + (size_t)N * 30;

  // ---- initial state (uniform across lanes) ----
  float Rot[9], v[3], p[3], bo[3], ba[3], Rci[9], tci[3];
  {
    float cr = cosf(ang0[0]), sr = sinf(ang0[0]);
    float cp = cosf(ang0[1]), sp = sinf(ang0[1]);
    float cy = cosf(ang0[2]), sy = sinf(ang0[2]);
    float rx[9] = {1, 0, 0, 0, cr, -sr, 0, sr, cr};
    float ry[9] = {cp, 0, sp, 0, 1, 0, -sp, 0, cp};
    float rz[9] = {cy, -sy, 0, sy, cy, 0, 0, 0, 1};
    float tmp[9];
    m3mul(rz, ry, tmp);
    m3mul(tmp, rx, Rot);
  }
#pragma unroll
  for (int k = 0; k < 3; ++k) {
    v[k] = vmes[k];
    p[k] = 0.f; bo[k] = 0.f; ba[k] = 0.f; tci[k] = 0.f;
  }
#pragma unroll
  for (int k = 0; k < 9; ++k) Rci[k] = 0.f;
  Rci[0] = Rci[4] = Rci[8] = 1.f;

  for (int idx = lane; idx < 1024; idx += 32) Pl[idx] = 0.f;
  __syncthreads();
  if (lane == 0) {
    const float pd[21] = {1e-3f, 1e-3f, 0.f, 0.1f, 0.1f, 0.f, 0.f, 0.f, 0.f,
                          6e-3f, 6e-3f, 6e-3f, 4e-3f, 4e-3f, 4e-3f,
                          1e-6f, 1e-6f, 1e-6f, 5e-3f, 5e-3f, 5e-3f};
#pragma unroll
    for (int k = 0; k < 21; ++k) Pl[MM(k, k)] = pd[k];
    // row 0 of every output
#pragma unroll
    for (int k = 0; k < 9; ++k) {
      oRot[k] = Rot[k];
      oRci[k] = Rci[k];
    }
#pragma unroll
    for (int k = 0; k < 3; ++k) {
      oV[k] = v[k]; oP[k] = p[k]; oBo[k] = bo[k]; oBa[k] = ba[k]; oTci[k] = tci[k];
    }
  }
  __syncthreads();

  const float Qd[18] = {1e-3f, 1e-3f, 1e-3f, 1e-2f, 1e-2f, 1e-2f,
                        6e-9f, 6e-9f, 6e-9f, 2e-4f, 2e-4f, 2e-4f,
                        1e-9f, 1e-9f, 1e-9f, 1e-9f, 1e-9f, 1e-9f};

  for (int i = 1; i < N; ++i) {
    const float dt = t[i] - t[i - 1];
    float ui[6];
#pragma unroll
    for (int k = 0; k < 6; ++k) ui[k] = u[(size_t)i * 6 + k];
    const float rm0 = mc[(size_t)i * 2 + 0];
    const float rm1 = mc[(size_t)i * 2 + 1];

    // ---- state propagation ----
    float am[3] = {ui[3] - ba[0], ui[4] - ba[1], ui[5] - ba[2]};
    float accv[3];
    m3v(Rot, am, accv);
    accv[2] -= 9.80665f;
    float vn[3], pn[3];
#pragma unroll
    for (int k = 0; k < 3; ++k) {
      vn[k] = v[k] + accv[k] * dt;
      pn[k] = p[k] + v[k] * dt + 0.5f * accv[k] * dt * dt;
    }
    float w[3] = {(ui[0] - bo[0]) * dt, (ui[1] - bo[1]) * dt, (ui[2] - bo[2]) * dt};
    float dRm[9];
    so3exp(w, dRm);
    float Rn[9];
    m3mul(Rot, dRm, Rn);

    float sv[9], sp_[9], vsr[9], psr[9];
    skew3(v, sv);
    skew3(p, sp_);
    m3mul(sv, Rot, vsr);
    m3mul(sp_, Rot, psr);

    // ---- F*dt -> bufA ----
    for (int idx = lane; idx < 1024; idx += 32) bufA[idx] = 0.f;
    __syncthreads();
    if (lane == 0) {
      bufA[MM(3, 1)] = 9.80665f * dt;   // skew(G_VEC)*dt
      bufA[MM(4, 0)] = -9.80665f * dt;
#pragma unroll
      for (int k = 0; k < 3; ++k) bufA[MM(6 + k, 3 + k)] = dt;
#pragma unroll
      for (int ii = 0; ii < 3; ++ii)
#pragma unroll
        for (int jj = 0; jj < 3; ++jj) {
          bufA[MM(ii, 9 + jj)] = -Rot[ii * 3 + jj] * dt;
          bufA[MM(3 + ii, 9 + jj)] = -vsr[ii * 3 + jj] * dt;
          bufA[MM(6 + ii, 9 + jj)] = -psr[ii * 3 + jj] * dt;
          bufA[MM(3 + ii, 12 + jj)] = -Rot[ii * 3 + jj] * dt;
        }
    }
    __syncthreads();
    mm32(bufA, bufA, bufB, lane);  // F^2
    __syncthreads();
    mm32(bufB, bufA, bufC, lane);  // F^2 @ F
    __syncthreads();
    // Phi -> bufD
    for (int idx = lane; idx < 1024; idx += 32) {
      int ii = idx >> 5, jj = idx & 31;
      bufD[idx] = ((ii == jj) ? 1.f : 0.f) + bufA[idx] + 0.5f * bufB[idx] +
                  (1.f / 6.f) * bufC[idx];
    }
    __syncthreads();
    // G*Q*dt -> bufA , (G*dt)^T -> bufB
    for (int idx = lane; idx < 1024; idx += 32) {
      bufA[idx] = 0.f;
      bufB[idx] = 0.f;
    }
    __syncthreads();
    if (lane == 0) {
#define PUTG(r, c, val)                     \
  do {                                      \
    float _v = (val);                       \
    bufA[MM((r), (c))] = _v * Qd[(c)];      \
    bufB[MM((c), (r))] = _v;                \
  } while (0)
#pragma unroll
      for (int ii = 0; ii < 3; ++ii)
#pragma unroll
        for (int jj = 0; jj < 3; ++jj) {
          PUTG(ii, jj, Rot[ii * 3 + jj] * dt);
          PUTG(3 + ii, jj, vsr[ii * 3 + jj] * dt);
          PUTG(6 + ii, jj, psr[ii * 3 + jj] * dt);
          PUTG(3 + ii, 3 + jj, Rot[ii * 3 + jj] * dt);
        }
#pragma unroll
      for (int k = 0; k < 3; ++k) {
        PUTG(9 + k, 6 + k, dt);
        PUTG(12 + k, 9 + k, dt);
        PUTG(15 + k, 12 + k, dt);
        PUTG(18 + k, 15 + k, dt);
      }
#undef PUTG
    }
    __syncthreads();
    mm32(bufA, bufB, bufC, lane);  // G Q G^T * dt^2
    __syncthreads();
    for (int idx = lane; idx < 1024; idx += 32) bufA[idx] = Pl[idx] + bufC[idx];
    __syncthreads();
    mm32(bufD, bufA, bufB, lane);  // Phi @ (P + GQG^T)
    __syncthreads();
    for (int idx = lane; idx < 1024; idx += 32) {
      int ii = idx >> 5, jj = idx & 31;
      bufC[idx] = bufD[jj * 32 + ii];  // Phi^T
    }
    __syncthreads();
    mm32(bufB, bufC, Pl, lane);  // P_n
    __syncthreads();

    // ---- measurement model ----
    float Rb[9];
    m3mul(Rn, Rci, Rb);
    float vimu[3];
    m3Tv(Rn, vn, vimu);
    float om[3] = {ui[0] - bo[0], ui[1] - bo[1], ui[2] - bo[2]};
    float svi[9];
    skew3(vimu, svi);
    float M1[9];
    m3Tmul(Rci, svi, M1);
    float stc[9];
    skew3(tci, stc);
    float som[9];
    skew3(om, som);
    float vb[3];
    {
      float t1[3], t2[3];
      m3Tv(Rci, vimu, t1);
      m3v(stc, om, t2);
#pragma unroll
      for (int k = 0; k < 3; ++k) vb[k] = t1[k] + t2[k];
    }
    const float r0 = -vb[1], r1 = -vb[2];

    if (lane == 0) {
#pragma unroll
      for (int c = 0; c < 21; ++c) {
        Hld[0][c] = 0.f;
        Hld[1][c] = 0.f;
      }
#pragma unroll
      for (int rr = 0; rr < 2; ++rr)
#pragma unroll
        for (int jj = 0; jj < 3; ++jj) {
          Hld[rr][3 + jj] = Rb[jj * 3 + (1 + rr)];     // Rot_body^T rows 1,2
          Hld[rr][15 + jj] = M1[(1 + rr) * 3 + jj];
          Hld[rr][9 + jj] = stc[(1 + rr) * 3 + jj];
          Hld[rr][18 + jj] = -som[(1 + rr) * 3 + jj];
        }
    }
    __syncthreads();
    if (lane < 21) {
      float a0 = 0.f, a1 = 0.f;
      for (int j = 0; j < 21; ++j) {
        float pv = Pl[MM(lane, j)];
        a0 += pv * Hld[0][j];
        a1 += pv * Hld[1][j];
      }
      PHtld[lane][0] = a0;  // (P_n H^T)[lane][*]  (row-of-P convention)
      PHtld[lane][1] = a1;
    }
    __syncthreads();
    float S00 = rm0, S01 = 0.f, S10 = 0.f, S11 = rm1;
    for (int c = 0; c < 21; ++c) {
      float h0 = Hld[0][c], h1 = Hld[1][c];
      S00 += h0 * PHtld[c][0];
      S01 += h0 * PHtld[c][1];
      S10 += h1 * PHtld[c][0];
      S11 += h1 * PHtld[c][1];
    }
    const float inv = 1.f / (S00 * S11 - S01 * S10);
    const float Si00 = S11 * inv, Si01 = -S01 * inv;
    const float Si10 = -S10 * inv, Si11 = S00 * inv;
    if (lane < 21) {
      float k0 = Si00 * PHtld[lane][0] + Si01 * PHtld[lane][1];
      float k1 = Si10 * PHtld[lane][0] + Si11 * PHtld[lane][1];
      Kld[lane][0] = k0;
      Kld[lane][1] = k1;
      dxld[lane] = k0 * r0 + k1 * r1;
    }
    __syncthreads();
    float dx[21];
#pragma unroll
    for (int c = 0; c < 21; ++c) dx[c] = dxld[c];

    // ---- Joseph covariance update ----
    for (int idx = lane; idx < 1024; idx += 32) {
      int ii = idx >> 5, jj = idx & 31;
      float val = 0.f;
      if (ii < 21 && jj < 21)
        val = ((ii == jj) ? 1.f : 0.f) - Kld[ii][0] * Hld[0][jj] -
              Kld[ii][1] * Hld[1][jj];
      bufA[idx] = val;           // I - K H
      bufB[jj * 32 + ii] = val;  // (I - K H)^T
    }
    __syncthreads();
    mm32(bufA, Pl, bufC, lane);
    __syncthreads();
    mm32(bufC, bufB, Pl, lane);
    __syncthreads();
    for (int idx = lane; idx < 1024; idx += 32) {
      int ii = idx >> 5, jj = idx & 31;
      float add = 0.f;
      if (ii < 21 && jj < 21)
        add = Kld[ii][0] * Kld[jj][0] * rm0 + Kld[ii][1] * Kld[jj][1] * rm1;
      bufD[idx] = Pl[idx] + add;
    }
    __syncthreads();
    for (int idx = lane; idx < 1024; idx += 32) {
      int ii = idx >> 5, jj = idx & 31;
      Pl[idx] = 0.5f * (bufD[idx] + bufD[jj * 32 + ii]);
    }
    __syncthreads();

    // ---- state correction ----
    float dR[9], dx0[3], dx1[3];
    sen3exp(dx, dR, dx0, dx1);
    float Ru[9];
    m3mul(dR, Rn, Ru);
    float vu[3], pu[3];
    m3v(dR, vn, vu);
    m3v(dR, pn, pu);
#pragma unroll
    for (int k = 0; k < 3; ++k) {
      vu[k] += dx0[k];
      pu[k] += dx1[k];
      bo[k] += dx[9 + k];
      ba[k] += dx[12 + k];
      tci[k] += dx[18 + k];
    }
    float dRc[9];
    so3exp(dx + 15, dRc);
    float Rcn[9];
    m3mul(dRc, Rci, Rcn);
#pragma unroll
    for (int k = 0; k < 9; ++k) {
      Rci[k] = Rcn[k];
      Rot[k] = Ru[k];
    }
#pragma unroll
    for (int k = 0; k < 3; ++k) {
      v[k] = vu[k];
      p[k] = pu[k];
    }

    if (lane == 0) {
#pragma unroll
      for (int k = 0; k < 9; ++k) {
        oRot[(size_t)i * 9 + k] = Rot[k];
        oRci[(size_t)i * 9 + k] = Rci[k];
      }
#pragma unroll
      for (int k = 0; k < 3; ++k) {
        oV[(size_t)i * 3 + k] = v[k];
        oP[(size_t)i * 3 + k] = p[k];
        oBo[(size_t)i * 3 + k] = bo[k];
        oBa[(size_t)i * 3 + k] = ba[k];
        oTci[(size_t)i * 3 + k] = tci[k];
      }
    }
  }
}

extern "C" void kernel_launch(void* const* d_in, const int* in_sizes, int n_in,
                              void* d_out, int out_size, void* d_ws, size_t ws_size,
                              hipStream_t stream) {
  const float* t = (const float*)d_in[0];
  const float* u = (const float*)d_in[1];
  const float* mc = (const float*)d_in[2];
  const float* vmes = (const float*)d_in[3];
  const float* ang0 = (const float*)d_in[5];
  const int N = in_sizes[0];
  if (N <= 0) return;
  // Single wave owns the serial scan; WMMA requires wave32 with EXEC all ones.
  iekf_scan_kernel<<<dim3(1), dim3(32), 0, stream>>>(t, u, mc, vmes, ang0,
                                                     (float*)d_out, N);
}